// HGGNet_33079838114112
// MI455X (gfx1250) — compile-verified
//
#include <hip/hip_runtime.h>
#include <hip/hip_bf16.h>

// ---------------------------------------------------------------------------
// HGGNet pipeline for MI455X (gfx1250, wave32).
// Constants from the reference: B=8, N=4096 -> 2048 -> 512 -> 128, K=16.
// All shape parameters are compile-time template constants (powers of two)
// so index math is shifts/masks, and the WMMA C-loop fully resolves its
// C0-boundary branch at compile time.
// ---------------------------------------------------------------------------

#define B 8
#define KNN_K 16
#define KNN_BLOCK 128
#define FPS_BLOCK 512
#define GN_BLOCK 256

typedef __attribute__((ext_vector_type(2))) float v2f;
typedef __attribute__((ext_vector_type(8))) float v8f;

#if defined(__gfx1250__) && __has_builtin(__builtin_amdgcn_wmma_f32_16x16x4_f32)
#define HAVE_WMMA_F32 1
#else
#define HAVE_WMMA_F32 0
#endif

// --------------------------- input projection ------------------------------
// f0[b,o,n] = b_in[o] + sum_c w_in[o,c] * x[b,n,c]   (O=8, C=3)
template <int N>
__global__ void proj_kernel(const float* __restrict__ x,
                            const float* __restrict__ w_in,
                            const float* __restrict__ b_in,
                            float* __restrict__ f0) {
  int g = blockIdx.x * blockDim.x + threadIdx.x;     // B*N
  int n = g & (N - 1), b = g / N;
  float px = x[(size_t)(b * N + n) * 3 + 0];
  float py = x[(size_t)(b * N + n) * 3 + 1];
  float pz = x[(size_t)(b * N + n) * 3 + 2];
#pragma unroll
  for (int o = 0; o < 8; ++o) {
    float v = b_in[o] + w_in[o * 3 + 0] * px + w_in[o * 3 + 1] * py +
              w_in[o * 3 + 2] * pz;
    f0[((size_t)b * 8 + o) * N + n] = v;
  }
}

// -------------------------------- KNN --------------------------------------
// For each query keep the 16 nearest keys (squared distance). Nq,Nk are
// multiples of 128 so every block maps to one batch.
template <int Nq, int Nk>
__global__ void knn_kernel(const float* __restrict__ qc,
                           const float* __restrict__ kc,
                           int* __restrict__ idx_out) {
  __shared__ float sk[KNN_BLOCK * 3];
  int g = blockIdx.x * blockDim.x + threadIdx.x;     // B*Nq
  int q = g & (Nq - 1), b = g / Nq;
  float qx = qc[(size_t)(b * Nq + q) * 3 + 0];
  float qy = qc[(size_t)(b * Nq + q) * 3 + 1];
  float qz = qc[(size_t)(b * Nq + q) * 3 + 2];
  float bd[KNN_K];
  int bi[KNN_K];
#pragma unroll
  for (int j = 0; j < KNN_K; ++j) { bd[j] = 3.4e38f; bi[j] = 0; }

  for (int t0 = 0; t0 < Nk; t0 += KNN_BLOCK) {
    __syncthreads();
    int kk = t0 + threadIdx.x;
    sk[threadIdx.x * 3 + 0] = kc[(size_t)(b * Nk + kk) * 3 + 0];
    sk[threadIdx.x * 3 + 1] = kc[(size_t)(b * Nk + kk) * 3 + 1];
    sk[threadIdx.x * 3 + 2] = kc[(size_t)(b * Nk + kk) * 3 + 2];
    if (t0 + KNN_BLOCK < Nk)  // gfx1250 global_prefetch of next tile
      __builtin_prefetch(&kc[(size_t)(b * Nk + t0 + KNN_BLOCK + threadIdx.x) * 3], 0, 0);
    __syncthreads();
    for (int j = 0; j < KNN_BLOCK; ++j) {
      float dx = qx - sk[j * 3 + 0];
      float dy = qy - sk[j * 3 + 1];
      float dz = qz - sk[j * 3 + 2];
      float d = dx * dx + dy * dy + dz * dz;
      if (d < bd[KNN_K - 1]) {
        bd[KNN_K - 1] = d;
        bi[KNN_K - 1] = t0 + j;
#pragma unroll
        for (int s = KNN_K - 1; s >= 1; --s) {
          if (bd[s] < bd[s - 1]) {
            float td = bd[s]; bd[s] = bd[s - 1]; bd[s - 1] = td;
            int ti = bi[s];  bi[s] = bi[s - 1]; bi[s - 1] = ti;
          }
        }
      }
    }
  }
#pragma unroll
  for (int j = 0; j < KNN_K; ++j)
    idx_out[((size_t)(b * Nq) + q) * KNN_K + j] = bi[j];
}

// ---------------- fused graph-feature + GEMM via WMMA ----------------------
// y[b,o,n,k] = sum_c W[o,c] * F[c,n,k]
//   F[c,n,k] = (c < C0) ? xk[b,c,idx[b,n,k]] - xq[b,c,n] : xq[b,c-C0,n]
// One wave per 16(o) x 16(k) tile of one point n; C stepped by 4 with
// V_WMMA_F32_16X16X4_F32. grid = B*(O/16)*Nq / 4 blocks of 128 threads.
// C0 % 4 == 0 in every layer, so "c0 < C0" is a compile-time branch per
// unrolled iteration (covers both K rows c0+2h, c0+2h+1 of this lane).
template <int O, int Cin, int C0, int Nq, int Nk>
__global__ void __launch_bounds__(128)
gconv_wmma(const float* __restrict__ W, const float* __restrict__ xk,
           const float* __restrict__ xq, const int* __restrict__ idx,
           float* __restrict__ y) {
  constexpr int OT = O >> 4;
  int wave = (blockIdx.x * blockDim.x + threadIdx.x) >> 5;
  int lane = threadIdx.x & 31;
  int n = wave & (Nq - 1);
  int t = wave / Nq;
  int ot = t & (OT - 1);
  int b = t / OT;
  int r = lane & 15;      // A row / C-D column
  int h = lane >> 4;      // half-wave
  int o0 = ot << 4;
  int kx = r;             // neighbor slot (tile column)
  int nb = idx[((size_t)b * Nq + n) * KNN_K + kx];
  const float* xkb = xk + (size_t)b * C0 * Nk;
  const float* xqb = xq + (size_t)b * C0 * Nq;
  const float* wr = W + (size_t)(o0 + r) * Cin + 2 * h;  // this lane's A row

#if HAVE_WMMA_F32
  v8f acc = {};
#pragma unroll 8
  for (int c0 = 0; c0 < Cin; c0 += 4) {
    int ca = c0 + 2 * h;  // this lane's two K rows (ISA 16x4 A layout)
    v2f a, bv;
    a[0] = wr[c0];
    a[1] = wr[c0 + 1];
    if (c0 < C0) {        // compile-time per unrolled iteration
      bv[0] = xkb[(size_t)ca * Nk + nb] - xqb[(size_t)ca * Nq + n];
      bv[1] = xkb[(size_t)(ca + 1) * Nk + nb] - xqb[(size_t)(ca + 1) * Nq + n];
    } else {
      bv[0] = xqb[(size_t)(ca - C0) * Nq + n];
      bv[1] = xqb[(size_t)(ca + 1 - C0) * Nq + n];
    }
    acc = __builtin_amdgcn_wmma_f32_16x16x4_f32(false, a, false, bv, (short)0,
                                                acc, false, false);
  }
  float* yb = y + (((size_t)b * O + o0) * Nq + n) * KNN_K + kx;
#pragma unroll
  for (int v = 0; v < 8; ++v) {
    int row = v + 8 * h;  // ISA 16x16 f32 C/D layout
    yb[(size_t)row * Nq * KNN_K] = acc[v];
  }
#else
  float accs[8] = {0.f, 0.f, 0.f, 0.f, 0.f, 0.f, 0.f, 0.f};
  for (int c = 0; c < Cin; ++c) {
    float fv = (c < C0) ? (xkb[(size_t)c * Nk + nb] - xqb[(size_t)c * Nq + n])
                        : xqb[(size_t)(c - C0) * Nq + n];
#pragma unroll
    for (int v = 0; v < 8; ++v)
      accs[v] += W[(size_t)(o0 + v + 8 * h) * Cin + c] * fv;
  }
  float* yb = y + (((size_t)b * O + o0) * Nq + n) * KNN_K + kx;
#pragma unroll
  for (int v = 0; v < 8; ++v)
    yb[(size_t)(v + 8 * h) * Nq * KNN_K] = accs[v];
#endif
}

// ------------------------- group-norm statistics ----------------------------
// One block per (b, group). Group slab is contiguous: Cg*Nq*16 floats.
template <int Cg, int Nq>
__global__ void gn_reduce(const float* __restrict__ y,
                          float* __restrict__ stats) {
  __shared__ float ss[GN_BLOCK], sq[GN_BLOCK];
  constexpr size_t cnt = (size_t)Cg * Nq * KNN_K;
  const float* p = y + (size_t)blockIdx.x * cnt;
  float s = 0.f, q = 0.f;
  for (size_t i = threadIdx.x; i < cnt; i += GN_BLOCK) {
    float v = p[i];
    s += v;
    q += v * v;
  }
  ss[threadIdx.x] = s;
  sq[threadIdx.x] = q;
  __syncthreads();
#pragma unroll
  for (int t = GN_BLOCK / 2; t > 0; t >>= 1) {
    if ((int)threadIdx.x < t) {
      ss[threadIdx.x] += ss[threadIdx.x + t];
      sq[threadIdx.x] += sq[threadIdx.x + t];
    }
    __syncthreads();
  }
  if (threadIdx.x == 0) {
    constexpr float inv = 1.0f / (float)cnt;
    float mu = ss[0] * inv;
    float var = sq[0] * inv - mu * mu;
    stats[blockIdx.x * 2 + 0] = mu;
    stats[blockIdx.x * 2 + 1] = var;
  }
}

// ------------------ group-norm apply + leaky-relu + max_k -------------------
template <int O, int Cg, int Nq>
__global__ void gn_apply_max(const float* __restrict__ y,
                             const float* __restrict__ stats,
                             const float* __restrict__ gamma,
                             const float* __restrict__ beta,
                             float* __restrict__ out) {
  int g = blockIdx.x * blockDim.x + threadIdx.x;  // B*O*Nq
  int n = g & (Nq - 1);
  int t = g / Nq;
  int o = t & (O - 1);
  int b = t / O;
  int grp = b * 4 + o / Cg;
  float mu = stats[grp * 2 + 0];
  float var = stats[grp * 2 + 1];
  float sc = gamma[o] * rsqrtf(var + 1e-5f);
  float sh = beta[o] - mu * sc;
  const float* p = y + ((size_t)(b * O + o) * Nq + n) * KNN_K;
  float m = -3.4e38f;
#pragma unroll
  for (int k = 0; k < KNN_K; ++k) {
    float v = p[k] * sc + sh;
    v = (v > 0.f) ? v : 0.2f * v;
    m = fmaxf(m, v);
  }
  out[(size_t)(b * O + o) * Nq + n] = m;
}

// --------------------- farthest point sampling (sequential) -----------------
template <int N>
__global__ void fps_kernel(const float* __restrict__ coor, int m,
                           int* __restrict__ idx_out) {
  __shared__ float dist[N];
  __shared__ float rv[FPS_BLOCK];
  __shared__ int ri[FPS_BLOCK];
  __shared__ int s_last;
  int b = blockIdx.x, tid = threadIdx.x;
  const float* c = coor + (size_t)b * N * 3;
#pragma unroll
  for (int i = tid; i < N; i += FPS_BLOCK) dist[i] = 1e10f;
  if (tid == 0) {
    idx_out[(size_t)b * m] = 0;
    s_last = 0;
  }
  __syncthreads();
  for (int it = 1; it < m; ++it) {
    int last = s_last;
    float lx = c[last * 3 + 0], ly = c[last * 3 + 1], lz = c[last * 3 + 2];
    float bv = -1.f;
    int bidx = 0;
#pragma unroll
    for (int i = tid; i < N; i += FPS_BLOCK) {
      float dx = c[i * 3 + 0] - lx;
      float dy = c[i * 3 + 1] - ly;
      float dz = c[i * 3 + 2] - lz;
      float d = dx * dx + dy * dy + dz * dz;
      float nd = fminf(dist[i], d);
      dist[i] = nd;
      if (nd > bv) { bv = nd; bidx = i; }   // ascending i -> first max kept
    }
    rv[tid] = bv;
    ri[tid] = bidx;
    __syncthreads();
#pragma unroll
    for (int s = FPS_BLOCK / 2; s > 0; s >>= 1) {
      if (tid < s) {
        if (rv[tid + s] > rv[tid] ||
            (rv[tid + s] == rv[tid] && ri[tid + s] < ri[tid])) {
          rv[tid] = rv[tid + s];
          ri[tid] = ri[tid + s];
        }
      }
      __syncthreads();
    }
    if (tid == 0) {
      s_last = ri[0];
      idx_out[(size_t)b * m + it] = ri[0];
    }
    __syncthreads();
  }
}

// ------------------------------- gathers ------------------------------------
template <int m, int Nsrc>
__global__ void gather_coor(const float* __restrict__ src,
                            const int* __restrict__ idx,
                            float* __restrict__ dst) {
  int g = blockIdx.x * blockDim.x + threadIdx.x;  // B*m
  int j = g & (m - 1), b = g / m;
  int s = idx[(size_t)b * m + j];
  dst[((size_t)b * m + j) * 3 + 0] = src[((size_t)b * Nsrc + s) * 3 + 0];
  dst[((size_t)b * m + j) * 3 + 1] = src[((size_t)b * Nsrc + s) * 3 + 1];
  dst[((size_t)b * m + j) * 3 + 2] = src[((size_t)b * Nsrc + s) * 3 + 2];
}

template <int C, int m, int Nsrc>
__global__ void gather_feat(const float* __restrict__ src,
                            const int* __restrict__ idx,
                            float* __restrict__ dst) {
  int g = blockIdx.x * blockDim.x + threadIdx.x;  // B*C*m
  int j = g & (m - 1);
  int t = g / m;
  int c = t & (C - 1);
  int b = t / C;
  dst[g] = src[((size_t)b * C + c) * Nsrc + idx[(size_t)b * m + j]];
}

// ------------------------------- finalize -----------------------------------
// d_out = [out_coor (8,128,3) | out_f (8,128,256)]
__global__ void finalize_kernel(const float* __restrict__ coor3,
                                const float* __restrict__ s3,
                                float* __restrict__ out) {
  int g = blockIdx.x * blockDim.x + threadIdx.x;  // 3072 + 262144 threads
  if (g < B * 128 * 3) {
    out[g] = coor3[g];
  } else {
    int i = g - B * 128 * 3;
    int c = i & 255;
    int t = i >> 8;
    int n = t & 127;
    int b = t >> 7;
    out[g] = s3[((size_t)b * 256 + c) * 128 + n];
  }
}

// ---------------------------------------------------------------------------
extern "C" void kernel_launch(void* const* d_in, const int* in_sizes, int n_in,
                              void* d_out, int out_size, void* d_ws,
                              size_t ws_size, hipStream_t stream) {
  (void)in_sizes; (void)n_in; (void)out_size; (void)ws_size;
  const float* x    = (const float*)d_in[0];   // (8,4096,3)
  // d_in[1..3] = num0/num1/num2 (2048/512/128, compile-time constants here)
  const float* w_in = (const float*)d_in[4];
  const float* b_in = (const float*)d_in[5];
  const float* w1 = (const float*)d_in[6],  *g1 = (const float*)d_in[7],  *be1 = (const float*)d_in[8];
  const float* w2 = (const float*)d_in[9],  *g2 = (const float*)d_in[10], *be2 = (const float*)d_in[11];
  const float* w4 = (const float*)d_in[12], *g4 = (const float*)d_in[13], *be4 = (const float*)d_in[14];
  const float* w5 = (const float*)d_in[15], *g5 = (const float*)d_in[16], *be5 = (const float*)d_in[17];
  const float* w6 = (const float*)d_in[18], *g6 = (const float*)d_in[19], *be6 = (const float*)d_in[20];
  const float* w7 = (const float*)d_in[21], *g7 = (const float*)d_in[22], *be7 = (const float*)d_in[23];
  float* out = (float*)d_out;

  // ---- workspace layout (~89 MB) ----
  char* wsp = (char*)d_ws;
  size_t off = 0;
  auto alloc = [&](size_t nbytes) -> char* {
    char* p = wsp + off;
    off += (nbytes + 255) & ~(size_t)255;
    return p;
  };
  float* f0    = (float*)alloc((size_t)B * 8 * 4096 * 4);
  float* ybuf  = (float*)alloc((size_t)16777216 * 4);   // max y: (8,64,2048,16)
  float* f1    = (float*)alloc((size_t)B * 32 * 4096 * 4);
  float* coor1 = (float*)alloc((size_t)B * 2048 * 3 * 4);
  float* fq1   = (float*)alloc((size_t)B * 32 * 2048 * 4);
  float* s1    = (float*)alloc((size_t)B * 64 * 2048 * 4);
  float* coor2 = (float*)alloc((size_t)B * 512 * 3 * 4);
  float* fq2   = (float*)alloc((size_t)B * 64 * 512 * 4);
  float* f4    = (float*)alloc((size_t)B * 128 * 512 * 4);
  float* s2    = (float*)alloc((size_t)B * 128 * 512 * 4);
  float* coor3 = (float*)alloc((size_t)B * 128 * 3 * 4);
  float* fq3   = (float*)alloc((size_t)B * 128 * 128 * 4);
  float* f6    = (float*)alloc((size_t)B * 256 * 128 * 4);
  float* s3    = (float*)alloc((size_t)B * 256 * 128 * 4);
  float* stats = (float*)alloc((size_t)B * 4 * 2 * 4);
  int*   idxk  = (int*)alloc((size_t)B * 4096 * KNN_K * 4);
  int*   idxf  = (int*)alloc((size_t)B * 2048 * 4);

  // ---- stage 0: input projection + self graph conv (O=32,Cin=16) ----
  proj_kernel<4096><<<B * 4096 / 256, 256, 0, stream>>>(x, w_in, b_in, f0);
  knn_kernel<4096, 4096><<<B * 4096 / KNN_BLOCK, KNN_BLOCK, 0, stream>>>(x, x, idxk);
  gconv_wmma<32, 16, 8, 4096, 4096><<<(B * 2 * 4096) / 4, 128, 0, stream>>>(w1, f0, f0, idxk, ybuf);
  gn_reduce<8, 4096><<<B * 4, GN_BLOCK, 0, stream>>>(ybuf, stats);
  gn_apply_max<32, 8, 4096><<<B * 32 * 4096 / 256, 256, 0, stream>>>(ybuf, stats, g1, be1, f1);

  // ---- downsample 4096 -> 2048 ----
  fps_kernel<4096><<<B, FPS_BLOCK, 0, stream>>>(x, 2048, idxf);
  gather_coor<2048, 4096><<<B * 2048 / 256, 256, 0, stream>>>(x, idxf, coor1);
  gather_feat<32, 2048, 4096><<<B * 32 * 2048 / 256, 256, 0, stream>>>(f1, idxf, fq1);

  // ---- stage 1: cross graph conv (O=64,Cin=64) ----
  knn_kernel<2048, 4096><<<B * 2048 / KNN_BLOCK, KNN_BLOCK, 0, stream>>>(coor1, x, idxk);
  gconv_wmma<64, 64, 32, 2048, 4096><<<(B * 4 * 2048) / 4, 128, 0, stream>>>(w2, f1, fq1, idxk, ybuf);
  gn_reduce<16, 2048><<<B * 4, GN_BLOCK, 0, stream>>>(ybuf, stats);
  gn_apply_max<64, 16, 2048><<<B * 64 * 2048 / 256, 256, 0, stream>>>(ybuf, stats, g2, be2, s1);

  // ---- downsample 2048 -> 512 ----
  fps_kernel<2048><<<B, FPS_BLOCK, 0, stream>>>(coor1, 512, idxf);
  gather_coor<512, 2048><<<B * 512 / 256, 256, 0, stream>>>(coor1, idxf, coor2);
  gather_feat<64, 512, 2048><<<B * 64 * 512 / 256, 256, 0, stream>>>(s1, idxf, fq2);

  // ---- stage 2a: cross graph conv (O=128,Cin=128) ----
  knn_kernel<512, 2048><<<B * 512 / KNN_BLOCK, KNN_BLOCK, 0, stream>>>(coor2, coor1, idxk);
  gconv_wmma<128, 128, 64, 512, 2048><<<(B * 8 * 512) / 4, 128, 0, stream>>>(w4, s1, fq2, idxk, ybuf);
  gn_reduce<32, 512><<<B * 4, GN_BLOCK, 0, stream>>>(ybuf, stats);
  gn_apply_max<128, 32, 512><<<B * 128 * 512 / 256, 256, 0, stream>>>(ybuf, stats, g4, be4, f4);

  // ---- stage 2b: self graph conv (O=128,Cin=256) ----
  knn_kernel<512, 512><<<B * 512 / KNN_BLOCK, KNN_BLOCK, 0, stream>>>(coor2, coor2, idxk);
  gconv_wmma<128, 256, 128, 512, 512><<<(B * 8 * 512) / 4, 128, 0, stream>>>(w5, f4, f4, idxk, ybuf);
  gn_reduce<32, 512><<<B * 4, GN_BLOCK, 0, stream>>>(ybuf, stats);
  gn_apply_max<128, 32, 512><<<B * 128 * 512 / 256, 256, 0, stream>>>(ybuf, stats, g5, be5, s2);

  // ---- downsample 512 -> 128 ----
  fps_kernel<512><<<B, FPS_BLOCK, 0, stream>>>(coor2, 128, idxf);
  gather_coor<128, 512><<<B * 128 / 256, 256, 0, stream>>>(coor2, idxf, coor3);
  gather_feat<128, 128, 512><<<B * 128 * 128 / 256, 256, 0, stream>>>(s2, idxf, fq3);

  // ---- stage 3a: cross graph conv (O=256,Cin=256) ----
  knn_kernel<128, 512><<<B * 128 / KNN_BLOCK, KNN_BLOCK, 0, stream>>>(coor3, coor2, idxk);
  gconv_wmma<256, 256, 128, 128, 512><<<(B * 16 * 128) / 4, 128, 0, stream>>>(w6, s2, fq3, idxk, ybuf);
  gn_reduce<64, 128><<<B * 4, GN_BLOCK, 0, stream>>>(ybuf, stats);
  gn_apply_max<256, 64, 128><<<B * 256 * 128 / 256, 256, 0, stream>>>(ybuf, stats, g6, be6, f6);

  // ---- stage 3b: self graph conv (O=256,Cin=512) ----
  knn_kernel<128, 128><<<B * 128 / KNN_BLOCK, KNN_BLOCK, 0, stream>>>(coor3, coor3, idxk);
  gconv_wmma<256, 512, 256, 128, 128><<<(B * 16 * 128) / 4, 128, 0, stream>>>(w7, f6, f6, idxk, ybuf);
  gn_reduce<64, 128><<<B * 4, GN_BLOCK, 0, stream>>>(ybuf, stats);
  gn_apply_max<256, 64, 128><<<B * 256 * 128 / 256, 256, 0, stream>>>(ybuf, stats, g7, be7, s3);

  // ---- outputs ----
  finalize_kernel<<<(B * 128 * 3 + B * 128 * 256) / 256, 256, 0, stream>>>(coor3, s3, out);
}